// SAGEEncoder_6854767805210
// MI455X (gfx1250) — compile-verified
//
#include <hip/hip_runtime.h>

// ---------------------------------------------------------------------------
// SAGEEncoder for MI455X (gfx1250, wave32)
//   layer: out = (scatter_mean(x[src]->dst)) @ Wl^T + x @ Wr^T + b
//   GEMM uses V_WMMA_F32_16X16X4_F32 (pure fp32 matrix op -> matches reference
//   numerics; GEMM is compute-trivial here, scatter phase is the roofline).
// ---------------------------------------------------------------------------

typedef float v2f __attribute__((ext_vector_type(2)));
typedef float v8f __attribute__((ext_vector_type(8)));

#define CH 128          // channels
#define LDST 132        // LDS row stride (pad: 132 % 64 = 4 -> conflict-free A reads)

// ---------------- zero workspace -------------------------------------------
__global__ void sage_zero(float* __restrict__ p, long n) {
    long i = (long)blockIdx.x * blockDim.x + threadIdx.x;
    long stride = (long)gridDim.x * blockDim.x;
    for (; i < n; i += stride) p[i] = 0.0f;
}

// ---------------- scatter-sum: one wave32 per edge -------------------------
__global__ __launch_bounds__(256) void sage_scatter(
    const float* __restrict__ x, const int* __restrict__ src,
    const int* __restrict__ dst, float* __restrict__ agg,
    float* __restrict__ deg, int nE, int addDeg)
{
    const int lane = threadIdx.x & 31;
    const int e = blockIdx.x * (blockDim.x >> 5) + (threadIdx.x >> 5);
    if (e >= nE) return;
    const int s = src[e];
    const int d = dst[e];
    // coalesced 512B gather of the source row (16B per lane)
    const float4 v = ((const float4*)(x + (size_t)s * CH))[lane];
    float* ap = agg + (size_t)d * CH + lane * 4;
    atomicAdd(ap + 0, v.x);
    atomicAdd(ap + 1, v.y);
    atomicAdd(ap + 2, v.z);
    atomicAdd(ap + 3, v.w);
    if (addDeg && lane == 0) atomicAdd(deg + d, 1.0f);
}

// ---------------- fused mean + dual GEMM + bias (+ReLU) --------------------
// block: 256 threads = 8 wave32; one block per 16-node row tile
// wave w computes output columns [16w, 16w+16) for those 16 nodes
__global__ __launch_bounds__(256) void sage_gemm(
    const float* __restrict__ aggsum, const float* __restrict__ deg,
    const float* __restrict__ xin,
    const float* __restrict__ Wl, const float* __restrict__ Wr,
    const float* __restrict__ bias, float* __restrict__ out,
    int nNodes, int relu)
{
    __shared__ float sA[16 * LDST];   // mean-normalized aggregate tile
    __shared__ float sX[16 * LDST];   // self-feature tile

    const int tid  = threadIdx.x;
    const int base = blockIdx.x * 16;               // first node of tile

    // ---- stage both tiles into LDS (float4 global loads, scalar LDS stores)
    for (int i = tid; i < 16 * CH / 4; i += 256) {
        const int row = (i * 4) >> 7;               // /128
        const int col = (i * 4) & (CH - 1);
        const int gr  = min(base + row, nNodes - 1);
        const float rdeg = 1.0f / fmaxf(deg[gr], 1.0f);
        const float4 va = ((const float4*)(aggsum + (size_t)gr * CH))[i & 31];
        const float4 vx = ((const float4*)(xin    + (size_t)gr * CH))[i & 31];
        float* pa = &sA[row * LDST + col];
        float* px = &sX[row * LDST + col];
        pa[0] = va.x * rdeg; pa[1] = va.y * rdeg; pa[2] = va.z * rdeg; pa[3] = va.w * rdeg;
        px[0] = vx.x;        px[1] = vx.y;        px[2] = vx.z;        px[3] = vx.w;
    }
    __syncthreads();

    const int wave  = tid >> 5;                     // 0..7 -> column tile
    const int lane  = tid & 31;
    const int nloc  = lane & 15;                    // N within tile (B/C/D layout)
    const int ncol  = wave * 16 + nloc;             // output channel
    const int khalf = (lane >> 4) << 1;             // 0 (lanes 0-15) or 2 (16-31)
    const int arowOff = nloc * LDST;                // A-fragment row (M = lane%16)
    const size_t ncolOff = (size_t)ncol * CH;       // W row (B[k][n] = W[n][k])

    // accumulator init = bias[n]  (C/D: VGPR j -> M=j+8*(lane>=16), N=lane%16)
    v8f c;
    const float bv = bias[ncol];
#pragma unroll
    for (int j = 0; j < 8; ++j) c[j] = bv;

#define SAGE_KLOOP(S, W)                                                      \
    _Pragma("unroll")                                                         \
    for (int k0 = 0; k0 < CH; k0 += 4) {                                      \
        v2f a, b;                                                             \
        a[0] = (S)[arowOff + k0 + khalf + 0];                                 \
        a[1] = (S)[arowOff + k0 + khalf + 1];                                 \
        b[0] = (W)[ncolOff + k0 + khalf + 0];                                 \
        b[1] = (W)[ncolOff + k0 + khalf + 1];                                 \
        c = __builtin_amdgcn_wmma_f32_16x16x4_f32(                            \
                false, a, false, b, (short)0, c, false, false);               \
    }

    SAGE_KLOOP(sA, Wl)   // mean-aggregate path
    SAGE_KLOOP(sX, Wr)   // self path
#undef SAGE_KLOOP

    if (relu) {
#pragma unroll
        for (int j = 0; j < 8; ++j) c[j] = fmaxf(c[j], 0.0f);
    }

    // store: row m = base + j + 8*(lane>=16), col = ncol
    const int rbase = base + ((lane >> 4) << 3);
#pragma unroll
    for (int j = 0; j < 8; ++j) {
        const int m = rbase + j;
        if (m < nNodes) out[(size_t)m * CH + ncol] = c[j];
    }
}

// ---------------------------------------------------------------------------
extern "C" void kernel_launch(void* const* d_in, const int* in_sizes, int n_in,
                              void* d_out, int out_size, void* d_ws, size_t ws_size,
                              hipStream_t stream)
{
    const float* x   = (const float*)d_in[0];
    const int*   ei  = (const int*)  d_in[1];   // [2, E] (src row, then dst row)
    const float* W1l = (const float*)d_in[2];
    const float* W1r = (const float*)d_in[3];
    const float* b1  = (const float*)d_in[4];
    const float* W2l = (const float*)d_in[5];
    const float* W2r = (const float*)d_in[6];
    const float* b2  = (const float*)d_in[7];
    float* out = (float*)d_out;

    const int N = in_sizes[0] / CH;     // 50000 nodes
    const int E = in_sizes[1] / 2;      // 800000 edges
    const int* src = ei;
    const int* dst = ei + E;

    float* agg = (float*)d_ws;                   // [N, 128]
    float* deg = agg + (size_t)N * CH;           // [N]
    float* h   = deg + N;                        // [N, 128]

    const int zeroBlocks = 2048;
    const int scatBlocks = (E + 7) / 8;          // 8 edges (waves) per block
    const int gemmBlocks = (N + 15) / 16;

    // -------- layer 1 --------
    sage_zero<<<zeroBlocks, 256, 0, stream>>>(agg, (long)N * (CH + 1)); // agg+deg
    sage_scatter<<<scatBlocks, 256, 0, stream>>>(x, src, dst, agg, deg, E, 1);
    sage_gemm<<<gemmBlocks, 256, 0, stream>>>(agg, deg, x, W1l, W1r, b1, h, N, 1);

    // -------- layer 2 (deg unchanged) --------
    sage_zero<<<zeroBlocks, 256, 0, stream>>>(agg, (long)N * CH);
    sage_scatter<<<scatBlocks, 256, 0, stream>>>(h, src, dst, agg, deg, E, 0);
    sage_gemm<<<gemmBlocks, 256, 0, stream>>>(agg, deg, h, W2l, W2r, b2, out, N, 0);
}